// LightGCN_34763465293828
// MI455X (gfx1250) — compile-verified
//
#include <hip/hip_runtime.h>
#include <hip/hip_bf16.h>
#include <math.h>

#define N_USERS   100000
#define N_ITEMS   50000
#define N_NODES   150000
#define EMBED_DIM 64
#define NNZ       2400000
#define BATCH     16384

typedef __attribute__((ext_vector_type(2))) float v2f;
typedef __attribute__((ext_vector_type(8))) float v8f;

// ---------------------------------------------------------------------------
// 1) x = acc = concat(user_embed, item_embed).  Grid sized exactly: no guard.
// ---------------------------------------------------------------------------
__global__ void lgcn_init(const float* __restrict__ ue, const float* __restrict__ ie,
                          float* __restrict__ x, float* __restrict__ acc) {
  int tid = blockIdx.x * blockDim.x + threadIdx.x;            // float4 index
  const int u4 = N_USERS * EMBED_DIM / 4;
  float4 v = (tid < u4) ? ((const float4*)ue)[tid]
                        : ((const float4*)ie)[tid - u4];
  ((float4*)x)[tid]   = v;
  ((float4*)acc)[tid] = v;
}

__global__ void lgcn_zero(float4* __restrict__ y) {
  int tid = blockIdx.x * blockDim.x + threadIdx.x;
  y[tid] = make_float4(0.f, 0.f, 0.f, 0.f);
}

// ---------------------------------------------------------------------------
// 2) Edge-parallel SpMM: y[row] += val * x[col].  16 threads per edge,
//    float4 per thread -> coalesced 256B gathers; unsafeAtomicAdd forces the
//    native non-returning global_atomic_add_f32 (L2-resident scatters:
//    38.4MB table << 192MB L2).  Grid exact (150000 x 256 = NNZ*16).
// ---------------------------------------------------------------------------
__global__ void lgcn_spmm(const float* __restrict__ vals,
                          const int* __restrict__ row,
                          const int* __restrict__ col,
                          const float* __restrict__ x,
                          float* __restrict__ y) {
  unsigned tid = blockIdx.x * blockDim.x + threadIdx.x;
  int e = (int)(tid >> 4);
  int s = (int)(tid & 15);
  float a = vals[e];
  int c = col[e], r = row[e];
  float4 xv = ((const float4*)(x + (size_t)c * EMBED_DIM))[s];
  float* out = y + (size_t)r * EMBED_DIM + s * 4;
  unsafeAtomicAdd(out + 0, a * xv.x);
  unsafeAtomicAdd(out + 1, a * xv.y);
  unsafeAtomicAdd(out + 2, a * xv.z);
  unsafeAtomicAdd(out + 3, a * xv.w);
}

__global__ void lgcn_accadd(float4* __restrict__ acc, const float4* __restrict__ y) {
  int tid = blockIdx.x * blockDim.x + threadIdx.x;
  float4 a = acc[tid], b = y[tid];
  a.x += b.x; a.y += b.y; a.z += b.z; a.w += b.w;
  acc[tid] = a;
}

// ---------------------------------------------------------------------------
// 3) BPR loss via V_WMMA_F32_16X16X4_F32.
//    One wave per 16 batch entries: D = U(16x64) x (P-N)^T(64x16); diagonal
//    of D is score_diff.  A/B frags per ISA layout: lane L holds row m=L&15,
//    K offsets {0,1} (L<16) or {2,3} (L>=16) as a float2.  Grid is exactly
//    1024 waves -> no divergence before WMMA (EXEC all 1s).
// ---------------------------------------------------------------------------
__global__ void lgcn_bpr_wmma(const float* __restrict__ acc,
                              const int* __restrict__ users,
                              const int* __restrict__ pos,
                              const int* __restrict__ neg,
                              float* __restrict__ sums) {
  int lane = threadIdx.x & 31;
  int wave = (blockIdx.x * blockDim.x + threadIdx.x) >> 5;   // 0..1023
  int m    = lane & 15;
  int koff = (lane >> 4) << 1;                               // 0 or 2
  int i    = wave * 16 + m;                                  // batch entry of this row

  const float* urow = acc + (size_t)users[i] * EMBED_DIM;
  const float* prow = acc + (size_t)(N_USERS + pos[i]) * EMBED_DIM;
  const float* nrow = acc + (size_t)(N_USERS + neg[i]) * EMBED_DIM;

  v8f c = {};
#pragma unroll
  for (int kk = 0; kk < EMBED_DIM; kk += 4) {
    v2f a = *(const v2f*)(urow + kk + koff);
    v2f p = *(const v2f*)(prow + kk + koff);
    v2f n = *(const v2f*)(nrow + kk + koff);
    v2f b = p - n;
    c = __builtin_amdgcn_wmma_f32_16x16x4_f32(false, a, false, b,
                                              (short)0, c, false, false);
  }

  // Diagonal: lanes 0..7 -> c[lane] (M=N=lane); lanes 24..31 -> c[lane-24]
  float diff = 0.f;
#pragma unroll
  for (int r = 0; r < 8; ++r) {
    if (lane == r)      diff = c[r];
    if (lane == 24 + r) diff = c[r];
  }

  float contrib = 0.f;
  if (lane < 8 || lane >= 24) {
    float xs  = diff * 0.0625f;                 // fold (1/4)*(1/4) layer average
    float sig = 1.0f / (1.0f + expf(-xs));
    contrib   = logf(sig + 1e-8f);
  }
#pragma unroll
  for (int off = 16; off; off >>= 1) contrib += __shfl_xor(contrib, off);
  if (lane == 0)
    __hip_atomic_fetch_add(&sums[0], contrib, __ATOMIC_RELAXED, __HIP_MEMORY_SCOPE_AGENT);
}

// ---------------------------------------------------------------------------
// 4) Regularizer: sum of squares of layer-0 gathered embeddings.
//    Grid exact (1024 x 256 = BATCH*16).
// ---------------------------------------------------------------------------
__global__ void lgcn_reg(const float* __restrict__ ue, const float* __restrict__ ie,
                         const int* __restrict__ users, const int* __restrict__ pos,
                         const int* __restrict__ neg, float* __restrict__ sums) {
  int tid = blockIdx.x * blockDim.x + threadIdx.x;
  int i = tid >> 4, s = tid & 15;
  float4 u = ((const float4*)(ue + (size_t)users[i] * EMBED_DIM))[s];
  float4 p = ((const float4*)(ie + (size_t)pos[i]   * EMBED_DIM))[s];
  float4 n = ((const float4*)(ie + (size_t)neg[i]   * EMBED_DIM))[s];
  float ss = u.x*u.x + u.y*u.y + u.z*u.z + u.w*u.w
           + p.x*p.x + p.y*p.y + p.z*p.z + p.w*p.w
           + n.x*n.x + n.y*n.y + n.z*n.z + n.w*n.w;
#pragma unroll
  for (int off = 16; off; off >>= 1) ss += __shfl_xor(ss, off);
  if ((threadIdx.x & 31) == 0)
    __hip_atomic_fetch_add(&sums[1], ss, __ATOMIC_RELAXED, __HIP_MEMORY_SCOPE_AGENT);
}

__global__ void lgcn_zero_sums(float* __restrict__ sums) {
  if (threadIdx.x < 2) sums[threadIdx.x] = 0.f;
}

__global__ void lgcn_finalize(const float* __restrict__ sums, float* __restrict__ out) {
  if (threadIdx.x == 0) {
    out[0] = -sums[0] / (float)BATCH;   // loss
    out[1] =  sums[1] / (float)BATCH;   // reg_loss
  }
}

// ---------------------------------------------------------------------------
extern "C" void kernel_launch(void* const* d_in, const int* in_sizes, int n_in,
                              void* d_out, int out_size, void* d_ws, size_t ws_size,
                              hipStream_t stream) {
  const float* ue   = (const float*)d_in[0];
  const float* ie   = (const float*)d_in[1];
  const float* vals = (const float*)d_in[2];
  const int*   row  = (const int*)d_in[3];
  const int*   col  = (const int*)d_in[4];
  const int*   usr  = (const int*)d_in[5];
  const int*   pit  = (const int*)d_in[6];
  const int*   nit  = (const int*)d_in[7];
  float* out = (float*)d_out;

  // Workspace layout: [0..63] scalar accumulators/pad, then acc, x, y buffers.
  float* W   = (float*)d_ws;
  float* acc = W + 64;
  float* bx  = acc + (size_t)N_NODES * EMBED_DIM;
  float* by  = bx  + (size_t)N_NODES * EMBED_DIM;

  const int TB = 256;
  const int gElem = (N_NODES * EMBED_DIM / 4) / TB;               // 9375 (exact)
  const int gSpmm = (int)(((long long)NNZ * 16) / TB);            // 150000 (exact)

  lgcn_zero_sums<<<1, 64, 0, stream>>>(W);
  lgcn_init<<<gElem, TB, 0, stream>>>(ue, ie, bx, acc);

  float* x = bx;
  float* y = by;
  for (int layer = 0; layer < 3; ++layer) {
    lgcn_zero<<<gElem, TB, 0, stream>>>((float4*)y);
    lgcn_spmm<<<gSpmm, TB, 0, stream>>>(vals, row, col, x, y);
    lgcn_accadd<<<gElem, TB, 0, stream>>>((float4*)acc, (const float4*)y);
    float* t = x; x = y; y = t;
  }

  // 1024 waves = 128 blocks x 256 threads (8 waves/block), exactly BATCH/16 tiles
  lgcn_bpr_wmma<<<BATCH / 16 / 8, TB, 0, stream>>>(acc, usr, pit, nit, W);
  lgcn_reg<<<(BATCH * 16) / TB, TB, 0, stream>>>(ue, ie, usr, pit, nit, W);
  lgcn_finalize<<<1, 64, 0, stream>>>(W, out);
}